// SparseConvNeXtLayerNorm_1726576857584
// MI455X (gfx1250) — compile-verified
//
#include <hip/hip_runtime.h>

// Sparse ConvNeXt LayerNorm for MI455X (gfx1250, wave32).
//
// Roofline: 154 MB in + 154 MB out, ~6 flops/elem -> pure HBM bandwidth.
// Mask-inactive pixels (~50%) skip the read entirely -> ~231 MB @ 23.3 TB/s.
// One wave32 per pixel: 384 ch = 12 floats/lane = 3 x b128 NT loads/stores.
//
// CDNA5 paths:
//   * tensor_load_to_lds (TDM D# descriptor) + s_wait_tensorcnt  -> weight/bias
//   * global_load_async_to_lds_b128 + s_wait_asynccnt            -> mask
//   * global_prefetch_b8 (__builtin_prefetch)                    -> next pixel
//   * wave32 butterfly reduction, scalar-ized uniform index math

typedef float        f4    __attribute__((ext_vector_type(4)));
typedef unsigned int u32x4 __attribute__((ext_vector_type(4)));
typedef unsigned int u32x8 __attribute__((ext_vector_type(8)));

namespace {
constexpr int   kC    = 384;
constexpr int   kCV4  = kC / 4;      // 96 f4 per pixel
constexpr int   kH    = 56;
constexpr int   kW    = 56;
constexpr int   kMH   = 7;
constexpr int   kMW   = 7;
constexpr int   kHRep = kH / kMH;    // 8
constexpr int   kWRep = kW / kMW;    // 8
constexpr float kEps  = 1e-6f;
}

// Issue one TDM copy of a contiguous row of `nelem` f32 from global to LDS.
// D# group0: count=1, lds_addr, global_addr[56:0], type=2 ("image").
// D# group1: data_size=4B, tensor_dim0 = tile_dim0 = nelem, single row.
__device__ __forceinline__ void tdm_load_row_f32(const void* gsrc,
                                                 unsigned int lds_off,
                                                 unsigned int nelem) {
  unsigned long long ga = (unsigned long long)gsrc;
  u32x4 g0 = { 1u,                                   // count=1, user desc
               lds_off,                              // lds_addr [63:32]
               (unsigned int)ga,                     // global_addr [95:64]
               ((unsigned int)(ga >> 32) & 0x01FFFFFFu) | (2u << 30) }; // +type=2
  u32x8 g1 = { 0x00020000u,                          // wg_mask=0, data_size=4B
               (nelem & 0xFFFFu) << 16,              // tensor_dim0[15:0] @63:48
               (nelem >> 16) | (1u << 16),           // tensor_dim0[31:16], tensor_dim1=1
               (nelem & 0xFFFFu) << 16,              // tile_dim0 @127:112
               1u,                                   // tile_dim1=1, tile_dim2=0
               nelem,                                // tensor_dim0_stride[31:0]
               0u, 0u };
  asm volatile("tensor_load_to_lds %0, %1" :: "s"(g0), "s"(g1) : "memory");
}

__global__ __launch_bounds__(256, 8)
void sparse_convnext_ln(const float* __restrict__ x,
                        const float* __restrict__ weight,
                        const float* __restrict__ bias,
                        const unsigned char* __restrict__ mask,
                        float* __restrict__ out,
                        int npix, int nmask16) {
  __shared__ float         wb[2 * kC];       // [0,384): weight  [384,768): bias
  __shared__ unsigned char lmask[32 * kMH * kMW];  // 1568 B

  const int tid  = threadIdx.x;
  const int lane = tid & 31;

  // ---- TDM: stage weight + bias rows into LDS (wave 0 issues; EXEC ignored
  //      by tensor ops, branch prevents duplicate issue from other waves). ----
  if (tid < 32) {
    const unsigned int wb_lds = (unsigned int)(size_t)(&wb[0]);
    tdm_load_row_f32(weight, wb_lds, kC);
    tdm_load_row_f32(bias, wb_lds + kC * sizeof(float), kC);
    __builtin_amdgcn_s_wait_tensorcnt(0);
  }

  // ---- Async global->LDS: stage the (B,7,7) bool mask (98 x b128). ----
  if (tid < nmask16) {
    const f4* gsrc = reinterpret_cast<const f4*>(mask) + tid;
    unsigned int lds_addr =
        (unsigned int)(size_t)(&lmask[0]) + (unsigned int)(tid * 16);
    asm volatile("global_load_async_to_lds_b128 %0, %1, off"
                 :
                 : "v"(lds_addr), "v"(gsrc)
                 : "memory");
  }
  asm volatile("s_wait_asynccnt 0x0" ::: "memory");
  __syncthreads();

  const f4* lw4 = reinterpret_cast<const f4*>(&wb[0]);
  const f4* lb4 = reinterpret_cast<const f4*>(&wb[kC]);

  // Force wave-uniformity so the pixel index math lands on the SALU and
  // x/out bases become SGPR pairs (SADDR-form loads/stores).
  const int wid =
      __builtin_amdgcn_readfirstlane((int)((blockIdx.x * blockDim.x + tid) >> 5));
  const int nwaves = (int)((gridDim.x * blockDim.x) >> 5);

  for (int p = wid; p < npix; p += nwaves) {
    const int b_ = p / (kH * kW);
    const int hw = p - b_ * (kH * kW);
    const int hh = hw / kW;
    const int ww = hw - hh * kW;
    const bool active =
        lmask[b_ * (kMH * kMW) + (hh / kHRep) * kMW + (ww / kWRep)] != 0;

    f4* op = reinterpret_cast<f4*>(out + (size_t)p * kC);

    if (!active) {
      // Masked pixel: stream zeros, never touch x (saves ~50% of read BW).
      const f4 z = (f4)0.0f;
      __builtin_nontemporal_store(z, op + lane);
      __builtin_nontemporal_store(z, op + lane + 32);
      __builtin_nontemporal_store(z, op + lane + 64);
      continue;
    }

    // Speculative prefetch of this wave's next pixel: 32 lanes x 48B stride
    // cover the full 1536B channel row.
    const int pn = p + nwaves;
    if (pn < npix)
      __builtin_prefetch(x + (size_t)pn * kC + lane * 12, 0, 0);

    const f4* xp = reinterpret_cast<const f4*>(x + (size_t)p * kC);
    const f4 v0 = __builtin_nontemporal_load(xp + lane);
    const f4 v1 = __builtin_nontemporal_load(xp + lane + 32);
    const f4 v2 = __builtin_nontemporal_load(xp + lane + 64);

    // Per-lane partial sum and sum-of-squares over 12 channels.
    float s = (v0.x + v0.y) + (v0.z + v0.w)
            + (v1.x + v1.y) + (v1.z + v1.w)
            + (v2.x + v2.y) + (v2.z + v2.w);
    float s2 = v0.x * v0.x;
    s2 = fmaf(v0.y, v0.y, s2); s2 = fmaf(v0.z, v0.z, s2);
    s2 = fmaf(v0.w, v0.w, s2); s2 = fmaf(v1.x, v1.x, s2);
    s2 = fmaf(v1.y, v1.y, s2); s2 = fmaf(v1.z, v1.z, s2);
    s2 = fmaf(v1.w, v1.w, s2); s2 = fmaf(v2.x, v2.x, s2);
    s2 = fmaf(v2.y, v2.y, s2); s2 = fmaf(v2.z, v2.z, s2);
    s2 = fmaf(v2.w, v2.w, s2);

    // wave32 butterfly reduction (5 steps, both accumulators together).
    #pragma unroll
    for (int off = 16; off > 0; off >>= 1) {
      s  += __shfl_xor(s,  off, 32);
      s2 += __shfl_xor(s2, off, 32);
    }

    const float mean = s * (1.0f / kC);
    float var = fmaf(-mean, mean, s2 * (1.0f / kC));
    var = fmaxf(var, 0.0f);
    const float rs = rsqrtf(var + kEps);

    const f4 w0 = lw4[lane], w1 = lw4[lane + 32], w2 = lw4[lane + 64];
    const f4 b0 = lb4[lane], b1 = lb4[lane + 32], b2 = lb4[lane + 64];

    const f4 o0 = (v0 - mean) * rs * w0 + b0;
    const f4 o1 = (v1 - mean) * rs * w1 + b1;
    const f4 o2 = (v2 - mean) * rs * w2 + b2;

    __builtin_nontemporal_store(o0, op + lane);
    __builtin_nontemporal_store(o1, op + lane + 32);
    __builtin_nontemporal_store(o2, op + lane + 64);
  }
}

extern "C" void kernel_launch(void* const* d_in, const int* in_sizes, int n_in,
                              void* d_out, int out_size, void* d_ws, size_t ws_size,
                              hipStream_t stream) {
  const float*         x      = (const float*)d_in[0];
  const float*         weight = (const float*)d_in[1];
  const float*         bias   = (const float*)d_in[2];
  const unsigned char* mask   = (const unsigned char*)d_in[3];  // jnp.bool_ -> 1 byte
  float*               out    = (float*)d_out;

  const int npix    = in_sizes[0] / kC;   // B*H*W = 100352
  const int nmask16 = in_sizes[3] / 16;   // 1568 B / 16 = 98 b128 copies

  // Persistent-style grid: 2048 blocks x 8 waves = 16384 waves,
  // ~6 pixels per wave -> LDS staging amortized, prefetch loop useful.
  const dim3 grid(2048);
  const dim3 block(256);
  hipLaunchKernelGGL(sparse_convnext_ln, grid, block, 0, stream,
                     x, weight, bias, mask, out, npix, nmask16);
}